// GGD_27608049778883
// MI455X (gfx1250) — compile-verified
//
#include <hip/hip_runtime.h>

#define DFEAT 128

typedef float v2f __attribute__((ext_vector_type(2)));
typedef float v8f __attribute__((ext_vector_type(8)));

__device__ __forceinline__ void atomic_add_f32(float* p, float v) {
    unsafeAtomicAdd(p, v);   // lowers to global_atomic_add_f32 on gfx1250
}

// ---------------------------------------------------------------- degree / norm
__global__ void deg_kernel(const int* __restrict__ dst, float* __restrict__ deg, int nE) {
    int i = blockIdx.x * blockDim.x + threadIdx.x;
    if (i < nE) atomic_add_f32(&deg[dst[i]], 1.0f);
}

__global__ void norm_kernel(const float* __restrict__ deg, float* __restrict__ nrm, int n) {
    int i = blockIdx.x * blockDim.x + threadIdx.x;
    if (i < n) nrm[i] = rsqrtf(fmaxf(deg[i], 1.0f));
}

// ------------------------------------------------- Wp row-sums + bias sum (tiny)
__global__ void wsum_kernel(const float* __restrict__ Wp, const float* __restrict__ bp,
                            float* __restrict__ wsum, float* __restrict__ bpsum) {
    int k = threadIdx.x;  // 128 threads
    float s = 0.0f;
#pragma unroll 8
    for (int j = 0; j < DFEAT; ++j) s += Wp[k * DFEAT + j];
    wsum[k] = s;
    __shared__ float red[DFEAT];
    red[k] = bp[k];
    __syncthreads();
    for (int off = DFEAT / 2; off > 0; off >>= 1) {
        if (k < off) red[k] += red[k + off];
        __syncthreads();
    }
    if (k == 0) *bpsum = red[0];
}

// ------------------------------------- gather src row, scale, scatter-add to dst
// one wave32 per edge; lane l covers features [4l .. 4l+3]
__global__ void scatter_kernel(const float* __restrict__ h, const float* __restrict__ nrm,
                               const int* __restrict__ src, const int* __restrict__ dst,
                               const int* __restrict__ perm, float* __restrict__ agg, int nE) {
    int lane = threadIdx.x & 31;
    int wid  = (blockIdx.x * blockDim.x + threadIdx.x) >> 5;
    int nW   = (gridDim.x * blockDim.x) >> 5;
    for (int e = wid; e < nE; e += nW) {
        int s = src[e];
        int d = dst[e];
        int row = perm ? perm[s] : s;
        float nv = nrm[s];
        int en = e + nW;                       // prefetch next edge's source row
        if (en < nE) {
            int sn = src[en];
            int rn = perm ? perm[sn] : sn;
            __builtin_prefetch(h + (size_t)rn * DFEAT + lane * 4, 0, 1);
        }
        float4 v = *(const float4*)(h + (size_t)row * DFEAT + lane * 4);
        float* a = agg + (size_t)d * DFEAT + lane * 4;
        atomic_add_f32(a + 0, v.x * nv);
        atomic_add_f32(a + 1, v.y * nv);
        atomic_add_f32(a + 2, v.z * nv);
        atomic_add_f32(a + 3, v.w * nv);
    }
}

// ---------------------------------------------------------------- WMMA fp32 GEMM
// out[r][c] = act( nrm[r] * (A @ W)[r][c] + bias[c] ), A:[nRows x 128], W:[128 x 128]
// one wave per 16x16 tile, V_WMMA_F32_16X16X4_F32 over K=128 (32 steps)
__global__ void __launch_bounds__(256)
gemm_kernel(const float* __restrict__ A, const float* __restrict__ nrm,
            const float* __restrict__ W, const float* __restrict__ bias,
            float* __restrict__ out, int nRows, int doRelu) {
    const int tilesN = DFEAT / 16;                      // 8
    int w = blockIdx.x * (blockDim.x >> 5) + (threadIdx.x >> 5);
    int tiles = (nRows / 16) * tilesN;
    if (w >= tiles) return;                             // wave-uniform exit; EXEC stays all-1
    int mt = w / tilesN, nt = w % tilesN;
    int lane = threadIdx.x & 31;
    int half = lane >> 4;                               // 0: lanes 0-15, 1: lanes 16-31
    int l16  = lane & 15;

    int arow = mt * 16 + l16;
    int bcol = nt * 16 + l16;
    const float* Arow = A + (size_t)arow * DFEAT;

    v8f acc = {};
#pragma unroll
    for (int kb = 0; kb < DFEAT; kb += 4) {
        // A 16x4 fp32 layout: VGPR0 -> K = kb + 2*half, VGPR1 -> K = kb + 2*half + 1
        v2f a = *(const v2f*)(Arow + kb + 2 * half);
        // B 4x16 fp32 layout (rows striped across lanes, row r & r+2 split by half)
        int k0 = kb + 2 * half;
        v2f b;
        b.x = W[(size_t)k0 * DFEAT + bcol];
        b.y = W[(size_t)(k0 + 1) * DFEAT + bcol];
        acc = __builtin_amdgcn_wmma_f32_16x16x4_f32(false, a, false, b,
                                                    (short)0, acc, false, false);
    }

    float bv = bias[bcol];
#pragma unroll
    for (int i = 0; i < 8; ++i) {
        // C/D layout: VGPR i holds M = i (lanes 0-15) and M = i+8 (lanes 16-31)
        int row = mt * 16 + half * 8 + i;
        float val = nrm[row] * acc[i] + bv;
        if (doRelu) val = fmaxf(val, 0.0f);
        out[(size_t)row * DFEAT + bcol] = val;
    }
}

// ----------------------------------- sc[n] = h2[n] . wsum + bpsum  (wave per node)
__global__ void score_kernel(const float* __restrict__ h, const float* __restrict__ wsum,
                             const float* __restrict__ bpsum, float* __restrict__ sc, int n) {
    int lane = threadIdx.x & 31;
    int w = blockIdx.x * (blockDim.x >> 5) + (threadIdx.x >> 5);
    if (w >= n) return;
    float4 hv = *(const float4*)(h + (size_t)w * DFEAT + lane * 4);
    float4 wv = *(const float4*)(wsum + lane * 4);
    float p = hv.x * wv.x + hv.y * wv.y + hv.z * wv.z + hv.w * wv.w;
#pragma unroll
    for (int off = 16; off > 0; off >>= 1) p += __shfl_xor(p, off, 32);
    if (lane == 0) sc[w] = p + bpsum[0];
}

// ------------------------ loss = mean( softplus(-sc1) ++ softplus(sc2) ) over 2N
__device__ __forceinline__ float softplus_f(float z) {
    return (z > 20.0f) ? z : log1pf(expf(z));
}

__global__ void loss_kernel(const float* __restrict__ sc1, const float* __restrict__ sc2,
                            float* __restrict__ out, int n) {
    int i = blockIdx.x * blockDim.x + threadIdx.x;
    float v = 0.0f;
    if (i < 2 * n) {
        float z = (i < n) ? -sc1[i] : sc2[i - n];
        v = softplus_f(z) * (1.0f / (float)(2 * n));
    }
    __shared__ float red[256];
    red[threadIdx.x] = v;
    __syncthreads();
    for (int off = 128; off > 0; off >>= 1) {
        if (threadIdx.x < off) red[threadIdx.x] += red[threadIdx.x + off];
        __syncthreads();
    }
    if (threadIdx.x == 0) atomic_add_f32(out, red[0]);
}

// ---------------------------------------------------------------------- driver
extern "C" void kernel_launch(void* const* d_in, const int* in_sizes, int n_in,
                              void* d_out, int out_size, void* d_ws, size_t ws_size,
                              hipStream_t stream) {
    const float* x   = (const float*)d_in[0];
    const float* W1  = (const float*)d_in[1];
    const float* b1  = (const float*)d_in[2];
    const float* W2  = (const float*)d_in[3];
    const float* b2  = (const float*)d_in[4];
    const float* Wp  = (const float*)d_in[5];
    const float* bp  = (const float*)d_in[6];
    const int* esrc  = (const int*)d_in[7];
    const int* edst  = (const int*)d_in[8];
    const int* perm  = (const int*)d_in[9];
    const int E = in_sizes[7];
    const int N = in_sizes[9];

    float* ws    = (float*)d_ws;
    float* deg   = ws;  ws += N;
    float* nrm   = ws;  ws += N;
    float* sc1   = ws;  ws += N;
    float* sc2   = ws;  ws += N;
    float* wsum  = ws;  ws += DFEAT;
    float* bpsum = ws;  ws += 16;                      // keep 16B alignment downstream
    float* agg   = ws;  ws += (size_t)N * DFEAT;
    float* h     = ws;

    hipMemsetAsync(d_out, 0, sizeof(float), stream);
    hipMemsetAsync(deg, 0, (size_t)N * sizeof(float), stream);

    deg_kernel <<<(E + 255) / 256, 256, 0, stream>>>(edst, deg, E);
    norm_kernel<<<(N + 255) / 256, 256, 0, stream>>>(deg, nrm, N);
    wsum_kernel<<<1, DFEAT, 0, stream>>>(Wp, bp, wsum, bpsum);

    const int gemmBlocks = ((N / 16) * (DFEAT / 16) + 7) / 8;   // 8 waves/block
    for (int br = 0; br < 2; ++br) {
        const int* p = br ? perm : nullptr;                     // corrupt: x[perm[src]]
        hipMemsetAsync(agg, 0, (size_t)N * DFEAT * sizeof(float), stream);
        scatter_kernel<<<2048, 256, 0, stream>>>(x, nrm, esrc, edst, p, agg, E);
        gemm_kernel<<<gemmBlocks, 256, 0, stream>>>(agg, nrm, W1, b1, h, N, 1);

        hipMemsetAsync(agg, 0, (size_t)N * DFEAT * sizeof(float), stream);
        scatter_kernel<<<2048, 256, 0, stream>>>(h, nrm, esrc, edst, nullptr, agg, E);
        gemm_kernel<<<gemmBlocks, 256, 0, stream>>>(agg, nrm, W2, b2, h, N, 0);

        score_kernel<<<(N + 7) / 8, 256, 0, stream>>>(h, wsum, bpsum, br ? sc2 : sc1, N);
    }

    loss_kernel<<<(2 * N + 255) / 256, 256, 0, stream>>>(sc1, sc2, (float*)d_out, N);
}